// Aligner_52896817218041
// MI455X (gfx1250) — compile-verified
//
#include <hip/hip_runtime.h>
#include <hip/hip_bf16.h>
#include <math.h>

typedef __attribute__((ext_vector_type(2))) float v2f;
typedef __attribute__((ext_vector_type(8))) float v8f;

#define BATCH 8
#define TT    256      // text timesteps
#define TM    1024     // mel timesteps
#define CCH   64       // channels (all conv outputs)
#define MEL   80       // mel input channels
#define TT3   250      // TT - 6
#define TM3   1018     // TM - 6
#define TTP   256      // padded text rows (multiple of 16)
#define TMP   1024     // padded mel rows  (multiple of 16)

// ---------------------------------------------------------------------------
// embed: e[b][t][c] = emb[x[b][t]][c], row-major [t][c], batch stride TMP*CCH
// ---------------------------------------------------------------------------
__global__ void embed_kernel(const int* __restrict__ x,
                             const float* __restrict__ emb,
                             float* __restrict__ out) {
    int i = blockIdx.x * blockDim.x + threadIdx.x;
    if (i >= BATCH * TT * CCH) return;
    int c = i & (CCH - 1);
    int t = (i >> 6) & (TT - 1);
    int b = i >> 14;
    out[b * (TMP * CCH) + t * CCH + c] = emb[x[b * TT + t] * CCH + c];
}

// ---------------------------------------------------------------------------
// repack all 6 conv weights [cout][cin][k] -> [cout][r], r = k*Cin + cin
// (k-major reduction index so WMMA B fragments are contiguous v2f loads)
// ---------------------------------------------------------------------------
struct RepackArgs {
    const float* src[6];
    float*       dst[6];
    int          cin[6];
};

__global__ void repack_kernel(RepackArgs a) {
    int layer = blockIdx.y;
    int Cin   = a.cin[layer];
    int ktot  = 3 * Cin;
    int n     = CCH * ktot;
    int i = blockIdx.x * blockDim.x + threadIdx.x;
    if (i >= n) return;
    int cout = i / ktot;
    int r    = i - cout * ktot;
    int k    = r / Cin;
    int ci   = r - k * Cin;
    a.dst[layer][i] = a.src[layer][cout * ktot + ci * 3 + k];
}

// ---------------------------------------------------------------------------
// conv1d (VALID, K=3, Cout=64) as im2col GEMM on WMMA f32 16x16x4.
//   out[t][cout] = bias[cout] + sum_r im2col[t][r] * Wt[cout][r]
//   r = k*CIN + cin  (CIN % 4 == 0 -> K=4 WMMA blocks never straddle a tap)
// Block: 128 thr = 4 waves; each wave = one 16-cout tile of 16 timesteps.
// Input tile (18 rows x CIN) staged in LDS, row stride CIN+2 (bank spread).
// A frag: lane(half,lrow): m=lrow, {A[m][kk],A[m][kk+1]} = contiguous LDS v2f
// B frag: {B[kk][n],B[kk+1][n]} = Wt[n*KTOT + kk..kk+1] contiguous global v2f
// TRANS=true reads raw mel [b][cin][t] directly (fused transpose).
// ---------------------------------------------------------------------------
template<int CIN, bool TRANS>
__global__ void __launch_bounds__(128)
conv_wmma_kernel(const float* __restrict__ in, const float* __restrict__ wt,
                 const float* __restrict__ bias, float* __restrict__ out,
                 int Tout, long in_bstride, int in_t_stride, int in_c_stride,
                 long out_bstride) {
    constexpr int KTOT = 3 * CIN;
    constexpr int SPAD = CIN + 2;
    __shared__ float sIn[18 * (MEL + 2)];

    int b   = blockIdx.y;
    int t0  = blockIdx.x * 16;
    int tid = threadIdx.x;
    int Tin = Tout + 2;

    for (int i = tid; i < 18 * CIN; i += 128) {
        int r, c;
        if (TRANS) { r = i % 18; c = i / 18; }        // coalesced along t
        else       { r = i / CIN; c = i % CIN; }      // coalesced along c
        int t = t0 + r;
        float v = (t < Tin)
            ? in[b * in_bstride + (long)t * in_t_stride + (long)c * in_c_stride]
            : 0.0f;
        sIn[r * SPAD + c] = v;
    }
    __syncthreads();

    int wave = tid >> 5;          // 0..3 -> cout tile
    int lane = tid & 31;
    int half = lane >> 4;
    int lrow = lane & 15;

    const float* wrow = wt + (wave * 16 + lrow) * KTOT;

    v8f acc = {};
#pragma unroll
    for (int k0 = 0; k0 < KTOT; k0 += 4) {
        int k  = k0 / CIN;                 // constant per unrolled iter
        int ci = (k0 % CIN) + half * 2;
        v2f a  = *(const v2f*)(&sIn[(lrow + k) * SPAD + ci]);
        v2f bw = *(const v2f*)(wrow + k0 + half * 2);
        acc = __builtin_amdgcn_wmma_f32_16x16x4_f32(
            false, a, false, bw, (short)0, acc, false, false);
    }

    float bv = bias[wave * 16 + lrow];     // n (cout) = lrow for both halves
#pragma unroll
    for (int v = 0; v < 8; ++v) {
        int t = t0 + half * 8 + v;         // C/D: M = v + 8*half
        if (t < Tout)
            out[b * out_bstride + (long)t * CCH + wave * 16 + lrow] = acc[v] + bv;
    }
}

// ---------------------------------------------------------------------------
__global__ void zero_kernel(float* __restrict__ p, int n) {
    int i = blockIdx.x * blockDim.x + threadIdx.x;
    if (i < n) p[i] = 0.0f;
}

// ---------------------------------------------------------------------------
// row norms: norm[row] = sum_c feat[row][c]^2
// ---------------------------------------------------------------------------
__global__ void norm_kernel(const float* __restrict__ feat,
                            float* __restrict__ norm) {
    __shared__ float s[64];
    int row = blockIdx.x;
    int c = threadIdx.x;
    float v = feat[row * CCH + c];
    s[c] = v * v;
    __syncthreads();
    for (int off = 32; off > 0; off >>= 1) {
        if (c < off) s[c] += s[c + off];
        __syncthreads();
    }
    if (c == 0) norm[row] = s[0];
}

// ---------------------------------------------------------------------------
// Pairwise L2 distance via ||t||^2 + ||m||^2 - 2 t.m, GEMM on WMMA f32.
// ---------------------------------------------------------------------------
__global__ void __launch_bounds__(32)
dist_wmma_kernel(const float* __restrict__ tF, const float* __restrict__ mF,
                 const float* __restrict__ tN, const float* __restrict__ mN,
                 float* __restrict__ out) {
    int b     = blockIdx.z;
    int ntile = blockIdx.x;
    int mtile = blockIdx.y;
    int lane  = threadIdx.x;
    int half  = lane >> 4;
    int lrow  = lane & 15;

    const float* arow = mF + (size_t)b * TMP * CCH + (size_t)(mtile * 16 + lrow) * CCH;
    const float* brow = tF + (size_t)b * TTP * CCH + (size_t)(ntile * 16 + lrow) * CCH;

    v8f acc = {};
#pragma unroll
    for (int k0 = 0; k0 < CCH; k0 += 4) {
        int kk = k0 + half * 2;
        v2f a  = *(const v2f*)(arow + kk);
        v2f bb = *(const v2f*)(brow + kk);
        acc = __builtin_amdgcn_wmma_f32_16x16x4_f32(
            false, a, false, bb, (short)0, acc, false, false);
    }

    int n = ntile * 16 + lrow;
    if (n >= TT3) return;
    float tn = tN[b * TTP + n];
#pragma unroll
    for (int v = 0; v < 8; ++v) {
        int m = mtile * 16 + half * 8 + v;
        if (m < TM3) {
            float mn = mN[b * TMP + m];
            float d2 = tn + mn - 2.0f * acc[v];
            out[(size_t)b * TM3 * TT3 + (size_t)m * TT3 + n] =
                -sqrtf(fmaxf(d2, 0.0f));
        }
    }
}

// ---------------------------------------------------------------------------
extern "C" void kernel_launch(void* const* d_in, const int* in_sizes, int n_in,
                              void* d_out, int out_size, void* d_ws, size_t ws_size,
                              hipStream_t stream) {
    (void)in_sizes; (void)n_in; (void)out_size; (void)ws_size;

    const int*   x   = (const int*)  d_in[0];
    const float* m   = (const float*)d_in[1];
    const float* emb = (const float*)d_in[2];
    const float* tw1 = (const float*)d_in[3];  const float* tb1 = (const float*)d_in[4];
    const float* tw2 = (const float*)d_in[5];  const float* tb2 = (const float*)d_in[6];
    const float* tw3 = (const float*)d_in[7];  const float* tb3 = (const float*)d_in[8];
    const float* mw1 = (const float*)d_in[9];  const float* mb1 = (const float*)d_in[10];
    const float* mw2 = (const float*)d_in[11]; const float* mb2 = (const float*)d_in[12];
    const float* mw3 = (const float*)d_in[13]; const float* mb3 = (const float*)d_in[14];
    float* out = (float*)d_out;

    // workspace layout (floats)
    float* ws   = (float*)d_ws;
    float* buf1 = ws;                          // B*1024*64
    float* buf2 = buf1 + BATCH * TMP * CCH;    // B*1024*64
    float* tF   = buf2 + BATCH * TMP * CCH;    // B*256*64   (padded, zeroed)
    float* mF   = tF   + BATCH * TTP * CCH;    // B*1024*64  (padded, zeroed)
    float* tN   = mF   + BATCH * TMP * CCH;    // B*256
    float* mN   = tN   + BATCH * TTP;          // B*1024
    float* wtT1 = mN   + BATCH * TMP;          // 64*192 each for text
    float* wtT2 = wtT1 + CCH * 3 * CCH;
    float* wtT3 = wtT2 + CCH * 3 * CCH;
    float* wtM1 = wtT3 + CCH * 3 * CCH;        // 64*240
    float* wtM2 = wtM1 + CCH * 3 * MEL;
    float* wtM3 = wtM2 + CCH * 3 * CCH;

    const long BSTR = (long)TMP * CCH;

    // zero padded feature buffers (tF and mF contiguous)
    int nz = BATCH * TTP * CCH + BATCH * TMP * CCH;
    zero_kernel<<<(nz + 255) / 256, 256, 0, stream>>>(tF, nz);

    // repack all 6 weight tensors in one launch
    RepackArgs ra;
    ra.src[0] = tw1; ra.dst[0] = wtT1; ra.cin[0] = CCH;
    ra.src[1] = tw2; ra.dst[1] = wtT2; ra.cin[1] = CCH;
    ra.src[2] = tw3; ra.dst[2] = wtT3; ra.cin[2] = CCH;
    ra.src[3] = mw1; ra.dst[3] = wtM1; ra.cin[3] = MEL;
    ra.src[4] = mw2; ra.dst[4] = wtM2; ra.cin[4] = CCH;
    ra.src[5] = mw3; ra.dst[5] = wtM3; ra.cin[5] = CCH;
    repack_kernel<<<dim3((CCH * 3 * MEL + 255) / 256, 6), 256, 0, stream>>>(ra);

    // ---- text branch (WMMA convs) ----
    embed_kernel<<<(BATCH * TT * CCH + 255) / 256, 256, 0, stream>>>(x, emb, buf1);
    conv_wmma_kernel<CCH, false><<<dim3((254 + 15) / 16, BATCH), 128, 0, stream>>>(
        buf1, wtT1, tb1, buf2, 254, BSTR, CCH, 1, BSTR);
    conv_wmma_kernel<CCH, false><<<dim3((252 + 15) / 16, BATCH), 128, 0, stream>>>(
        buf2, wtT2, tb2, buf1, 252, BSTR, CCH, 1, BSTR);
    conv_wmma_kernel<CCH, false><<<dim3((250 + 15) / 16, BATCH), 128, 0, stream>>>(
        buf1, wtT3, tb3, tF, 250, BSTR, CCH, 1, (long)TTP * CCH);

    // ---- mel branch (conv1 fuses the [C,T]->[T,C] transpose) ----
    conv_wmma_kernel<MEL, true><<<dim3((1022 + 15) / 16, BATCH), 128, 0, stream>>>(
        m, wtM1, mb1, buf1, 1022, (long)MEL * TM, 1, TM, BSTR);
    conv_wmma_kernel<CCH, false><<<dim3((1020 + 15) / 16, BATCH), 128, 0, stream>>>(
        buf1, wtM2, mb2, buf2, 1020, BSTR, CCH, 1, BSTR);
    conv_wmma_kernel<CCH, false><<<dim3((1018 + 15) / 16, BATCH), 128, 0, stream>>>(
        buf2, wtM3, mb3, mF, 1018, BSTR, CCH, 1, BSTR);

    // ---- row norms ----
    norm_kernel<<<BATCH * TTP, 64, 0, stream>>>(tF, tN);
    norm_kernel<<<BATCH * TMP, 64, 0, stream>>>(mF, mN);

    // ---- pairwise distance via WMMA GEMM ----
    dist_wmma_kernel<<<dim3(TTP / 16, TMP / 16, BATCH), 32, 0, stream>>>(
        tF, mF, tN, mN, out);
}